// MolecularGCN_87514253623368
// MI455X (gfx1250) — compile-verified
//
#include <hip/hip_runtime.h>
#include <hip/hip_bf16.h>

#define HID 64

typedef __attribute__((ext_vector_type(2))) float v2f;
typedef __attribute__((ext_vector_type(8))) float v8f;

// ---------------------------------------------------------------- utility ---
__global__ void k_zero(float* __restrict__ p, long n) {
  long i = (long)blockIdx.x * blockDim.x + threadIdx.x;
  const long st = (long)gridDim.x * blockDim.x;
  for (; i < n; i += st) p[i] = 0.0f;
}

// deg[dst] += 1 for each edge
__global__ void k_deg(const int* __restrict__ ei, float* __restrict__ deg, long E) {
  long i = (long)blockIdx.x * blockDim.x + threadIdx.x;
  const long st = (long)gridDim.x * blockDim.x;
  for (; i < E; i += st) unsafeAtomicAdd(deg + ei[E + i], 1.0f);
}

// deg -> rsqrt(deg + 1)   (self-loop included)
__global__ void k_rsqrt(float* __restrict__ d, int N) {
  int i = blockIdx.x * blockDim.x + threadIdx.x;
  if (i < N) d[i] = rsqrtf(d[i] + 1.0f);
}

// ------------------------------------------------------------- WMMA GEMM ----
// out[M, 64] = A[M, K] @ W[64, K]^T   (optionally +bias, ReLU)
// If agg != nullptr also writes agg = out * dis[row]^2  (GCN self-loop term).
// One wave computes a 16x64 tile via 4 accumulators of v_wmma_f32_16x16x4_f32.
template <int K>
__global__ __launch_bounds__(128) void k_gemm_wmma(
    const float* __restrict__ A, const float* __restrict__ W,
    const float* __restrict__ bias, const float* __restrict__ dis,
    float* __restrict__ out, float* __restrict__ agg, int M, int doRelu) {
  constexpr int KP = K + 4;  // pad LDS rows: stride 4 banks -> conflict-free
  __shared__ float sW[HID * KP];
  for (int i = threadIdx.x; i < HID * K; i += 128) {
    const int r = i / K, c = i - r * K;
    sW[r * KP + c] = W[i];
  }
  __syncthreads();

  const int lane  = threadIdx.x & 31;
  const int wave  = threadIdx.x >> 5;
  const int tileM = blockIdx.x * 4 + wave;
  if (tileM * 16 >= M) return;  // wave-uniform: EXEC stays all-1 for WMMA

  const int hi  = lane >> 4;   // 0: K-pair {0,1}, 1: K-pair {2,3}
  const int l15 = lane & 15;   // A: row-in-tile, B/C: column-in-tile
  const long arow = (long)(tileM * 16 + l15) * K;

  v8f acc[4];
#pragma unroll
  for (int nt = 0; nt < 4; ++nt) acc[nt] = 0.0f;

  for (int k0 = 0; k0 < K; k0 += 4) {
    const int kk = k0 + hi * 2;
    v2f a;
    a.x = A[arow + kk];
    a.y = A[arow + kk + 1];
#pragma unroll
    for (int nt = 0; nt < 4; ++nt) {
      const float* wp = sW + (nt * 16 + l15) * KP + kk;
      v2f b;
      b.x = wp[0];
      b.y = wp[1];
      acc[nt] = __builtin_amdgcn_wmma_f32_16x16x4_f32(
          false, a, false, b, (short)0, acc[nt], false, false);
    }
  }

  // C/D layout: VGPR r -> M = r (lanes 0-15) / r+8 (lanes 16-31), N = lane&15
#pragma unroll
  for (int r = 0; r < 8; ++r) {
    const int orow = tileM * 16 + hi * 8 + r;
    float dsq = 0.0f;
    if (agg) {
      const float ds = dis[orow];
      dsq = ds * ds;
    }
#pragma unroll
    for (int nt = 0; nt < 4; ++nt) {
      const int ocol = nt * 16 + l15;
      float v = acc[nt][r];
      if (bias) v += bias[ocol];
      if (doRelu) v = fmaxf(v, 0.0f);
      const long o = (long)orow * HID + ocol;
      out[o] = v;
      if (agg) agg[o] = v * dsq;
    }
  }
}

// ------------------------------------------------------ edge scatter-add ----
// agg[dst] += hw[src] * dis[src]*dis[dst]; 16 lanes per edge, float4 each.
__global__ void k_scatter(const int* __restrict__ ei, const float* __restrict__ dis,
                          const float* __restrict__ hw, float* __restrict__ agg,
                          long E) {
  long i = (long)blockIdx.x * blockDim.x + threadIdx.x;
  const long st = (long)gridDim.x * blockDim.x;
  const long nWork = E * 16;
  for (; i < nWork; i += st) {
    const long e = i >> 4;
    const int q = (int)(i & 15);
    const int s = ei[e];
    const int d = ei[E + e];
    const float norm = dis[s] * dis[d];
    const float4 v = *(const float4*)(hw + (long)s * HID + q * 4);
    float* dst = agg + (long)d * HID + q * 4;
    unsafeAtomicAdd(dst + 0, v.x * norm);
    unsafeAtomicAdd(dst + 1, v.y * norm);
    unsafeAtomicAdd(dst + 2, v.z * norm);
    unsafeAtomicAdd(dst + 3, v.w * norm);
  }
}

// h = relu(agg + b)
__global__ void k_bias_relu(const float* __restrict__ agg, const float* __restrict__ b,
                            float* __restrict__ h, long n) {
  long i = (long)blockIdx.x * blockDim.x + threadIdx.x;
  const long st = (long)gridDim.x * blockDim.x;
  for (; i < n; i += st) h[i] = fmaxf(agg[i] + b[i & (HID - 1)], 0.0f);
}

// -------------------------------------------------------------- pooling -----
__global__ void k_pool(const float* __restrict__ h, const int* __restrict__ batch,
                       float* __restrict__ pooled, float* __restrict__ cnts, int N) {
  long i = (long)blockIdx.x * blockDim.x + threadIdx.x;
  const long st = (long)gridDim.x * blockDim.x;
  const long n = (long)N * HID;
  for (; i < n; i += st) {
    const int node = (int)(i >> 6);
    const int j = (int)(i & (HID - 1));
    const int g = batch[node];
    unsafeAtomicAdd(pooled + (long)g * HID + j, h[i]);
    if (j == 0) unsafeAtomicAdd(cnts + g, 1.0f);
  }
}

// ------------------------------------------------------------- MLP head -----
__global__ __launch_bounds__(64) void k_head(
    const float* __restrict__ pooled, const float* __restrict__ cnts,
    const float* __restrict__ Wf1, const float* __restrict__ bf1,
    const float* __restrict__ Wf2, const float* __restrict__ bf2,
    float* __restrict__ out) {
  __shared__ float m[HID];
  __shared__ float t[HID];
  const int g = blockIdx.x;
  const int tid = threadIdx.x;
  const float cnt = fmaxf(cnts[g], 1.0f);
  m[tid] = pooled[(long)g * HID + tid] / cnt;
  __syncthreads();
  float a = bf1[tid];
  for (int k = 0; k < HID; ++k) a += m[k] * Wf1[tid * HID + k];
  t[tid] = fmaxf(a, 0.0f);
  __syncthreads();
  if (tid == 0) {
    float s = bf2[0];
    for (int o = 0; o < HID; ++o) s += t[o] * Wf2[o];
    out[g] = s;
  }
}

// ---------------------------------------------------------------- launch ----
extern "C" void kernel_launch(void* const* d_in, const int* in_sizes, int n_in,
                              void* d_out, int out_size, void* d_ws, size_t ws_size,
                              hipStream_t stream) {
  const float* x     = (const float*)d_in[0];
  const int*   ei    = (const int*)d_in[1];
  const int*   batch = (const int*)d_in[2];
  const float* W_enc = (const float*)d_in[3];
  const float* b_enc = (const float*)d_in[4];
  const float* Wl[3] = {(const float*)d_in[5], (const float*)d_in[7], (const float*)d_in[9]};
  const float* bl[3] = {(const float*)d_in[6], (const float*)d_in[8], (const float*)d_in[10]};
  const float* Wf1 = (const float*)d_in[11];
  const float* bf1 = (const float*)d_in[12];
  const float* Wf2 = (const float*)d_in[13];
  const float* bf2 = (const float*)d_in[14];
  float* out = (float*)d_out;

  const int  N = in_sizes[2];             // 100000 nodes
  const long E = (long)in_sizes[1] / 2;   // 3.2M edges
  const int  G = out_size;                // 1024 graphs

  // workspace layout (floats)
  float* ws = (float*)d_ws;
  const long NF    = (long)N * HID;
  const long oH    = ((long)N + 1023) / 1024 * 1024;
  float* dis    = ws;                     // [N]
  float* h      = ws + oH;                // [N,64]
  float* hw     = h + NF;                 // [N,64]
  float* agg    = hw + NF;                // [N,64]
  float* pooled = agg + NF;               // [G,64]
  float* cnts   = pooled + (long)G * HID; // [G]

  // degree + normalization
  k_zero<<<1024, 256, 0, stream>>>(dis, N);
  k_deg<<<2048, 256, 0, stream>>>(ei, dis, E);
  k_rsqrt<<<(N + 255) / 256, 256, 0, stream>>>(dis, N);

  const int tiles   = (N + 15) / 16;      // 6250, N divisible by 16
  const int gBlocks = (tiles + 3) / 4;

  // encoder: h = relu(x @ W_enc^T + b_enc)
  k_gemm_wmma<128><<<gBlocks, 128, 0, stream>>>(x, W_enc, b_enc, nullptr,
                                                h, nullptr, N, 1);

  for (int l = 0; l < 3; ++l) {
    // hw = h @ W^T ; agg = hw * dis^2 (self-loop) fused in epilogue
    k_gemm_wmma<64><<<gBlocks, 128, 0, stream>>>(h, Wl[l], nullptr, dis,
                                                 hw, agg, N, 0);
    k_scatter<<<8192, 256, 0, stream>>>(ei, dis, hw, agg, E);
    k_bias_relu<<<4096, 256, 0, stream>>>(agg, bl[l], h, NF);
  }

  // mean pool + head
  k_zero<<<256, 256, 0, stream>>>(pooled, (long)G * HID + G);  // pooled+cnts contiguous
  k_pool<<<8192, 256, 0, stream>>>(h, batch, pooled, cnts, N);
  k_head<<<G, 64, 0, stream>>>(pooled, cnts, Wf1, bf1, Wf2, bf2, out);
}